// CausalSelfAttention_73598559584679
// MI455X (gfx1250) — compile-verified
//
#include <hip/hip_runtime.h>
#include <hip/hip_bf16.h>

typedef __attribute__((ext_vector_type(2))) float v2f;
typedef __attribute__((ext_vector_type(8))) float v8f;

// D = A(16x4 f32) x B(4x16 f32) + C, f32 accum, full-wave (wave32) op.
__device__ __forceinline__ v8f wmma_f32(v2f a, v2f b, v8f c) {
  return __builtin_amdgcn_wmma_f32_16x16x4_f32(
      /*neg_a=*/false, a, /*neg_b=*/false, b,
      /*c_mod=*/(short)0, c, /*reuse_a=*/false, /*reuse_b=*/false);
}

// ---------------------------------------------------------------------------
// GEMM: C[M x N] = A[M x K] * B[N x K]^T   (all row-major, fp32)
// One wave per 32x64 output tile: 2 (rows) x 4 (cols) WMMA accumulators.
// Software-pipelined: fragments for k+4 are loaded before the 8 WMMAs of k.
// Scalar base + 32-bit element offsets -> global_load saddr form (no u64 VALU).
// ---------------------------------------------------------------------------
__global__ __launch_bounds__(32)
void gemm_nt_wmma(const float* __restrict__ A, const float* __restrict__ B,
                  float* __restrict__ C, int K, int lda, int ldb, int ldc) {
  const int lane = threadIdx.x & 31;
  const int half = lane >> 4;   // 0 or 1
  const int l16  = lane & 15;
  const int bm = blockIdx.y * 32;
  const int bn = blockIdx.x * 64;

  v8f acc[2][4];
#pragma unroll
  for (int i = 0; i < 2; ++i)
#pragma unroll
    for (int j = 0; j < 4; ++j) acc[i][j] = (v8f){};

  uint32_t aoff[2], boff[4];
#pragma unroll
  for (int i = 0; i < 2; ++i)
    aoff[i] = (uint32_t)(bm + i * 16 + l16) * (uint32_t)lda + 2u * half;
#pragma unroll
  for (int j = 0; j < 4; ++j)
    boff[j] = (uint32_t)(bn + j * 16 + l16) * (uint32_t)ldb + 2u * half;

  // prologue: fragments for k0 = 0
  v2f af[2], bf[4];
#pragma unroll
  for (int i = 0; i < 2; ++i) af[i] = *(const v2f*)(A + aoff[i]);
#pragma unroll
  for (int j = 0; j < 4; ++j) bf[j] = *(const v2f*)(B + boff[j]);

#pragma unroll 2
  for (int k0 = 0; k0 < K - 4; k0 += 4) {
    // issue next step's 6 loads first, then compute on current fragments
    v2f afn[2], bfn[4];
#pragma unroll
    for (int i = 0; i < 2; ++i)
      afn[i] = *(const v2f*)(A + aoff[i] + (uint32_t)(k0 + 4));
#pragma unroll
    for (int j = 0; j < 4; ++j)
      bfn[j] = *(const v2f*)(B + boff[j] + (uint32_t)(k0 + 4));
#pragma unroll
    for (int i = 0; i < 2; ++i)
#pragma unroll
      for (int j = 0; j < 4; ++j) acc[i][j] = wmma_f32(af[i], bf[j], acc[i][j]);
#pragma unroll
    for (int i = 0; i < 2; ++i) af[i] = afn[i];
#pragma unroll
    for (int j = 0; j < 4; ++j) bf[j] = bfn[j];
  }
  // epilogue step
#pragma unroll
  for (int i = 0; i < 2; ++i)
#pragma unroll
    for (int j = 0; j < 4; ++j) acc[i][j] = wmma_f32(af[i], bf[j], acc[i][j]);

#pragma unroll
  for (int i = 0; i < 2; ++i)
#pragma unroll
    for (int j = 0; j < 4; ++j) {
      const uint32_t row0 = (uint32_t)(bm + i * 16 + half * 8);
      const uint32_t col  = (uint32_t)(bn + j * 16 + l16);
#pragma unroll
      for (int v = 0; v < 8; ++v)
        C[(row0 + v) * (uint32_t)ldc + col] = acc[i][j][v];
    }
}

// ---------------------------------------------------------------------------
// RoPE in place on q and k halves of qkv[T x 3072]. One thread per (t, pair).
// ---------------------------------------------------------------------------
__global__ __launch_bounds__(256)
void rope_kernel(float* __restrict__ qkv, int T) {
  int idx = blockIdx.x * blockDim.x + threadIdx.x;   // over T * 1024
  if (idx >= T * 1024) return;
  int t = idx >> 10;
  int r = idx & 1023;
  int qksel = r >> 9;        // 0 = q, 1 = k
  int h = (r >> 5) & 15;
  int p = r & 31;
  float inv = powf(10000.0f, -(float)(2 * p) / 64.0f);
  float ang = (float)t * inv;
  float s = sinf(ang), c = cosf(ang);
  float* base = qkv + (uint32_t)t * 3072u + (uint32_t)(qksel * 1024 + h * 64 + 2 * p);
  float x1 = base[0], x2 = base[1];
  base[0] = x1 * c - x2 * s;
  base[1] = x2 * c + x1 * s;
}

// ---------------------------------------------------------------------------
// Flash attention (causal). 8 waves per block, each wave owns 16 query rows
// of one head; block covers 128 rows. Per 64-key tile:
//  - block cooperatively stages K (row-major) and V (transposed) into LDS
//    with stride-68 padding (conflict-free b64 fragment reads),
//  - S = Q K^T via WMMA, online softmax with shfl_xor row reductions,
//  - P staged via padded per-wave LDS slab into A-fragment layout,
//  - O += P V via WMMA with V B-fragments as single ds_load_b64.
// Next tile is prefetched (global_prefetch) while current tile computes.
// ---------------------------------------------------------------------------
#define LDT 68   // padded LDS row stride (floats): even (b64-aligned), bank-safe

__global__ __launch_bounds__(256)
void attn_kernel(const float* __restrict__ qkv, float* __restrict__ out) {
  __shared__ float Klds[64 * LDT];        // [key][c]
  __shared__ float Vlds[64 * LDT];        // transposed: [c][key]
  __shared__ float Plds[8 * 16 * LDT];    // per-wave 16-row P tiles

  const int tid  = threadIdx.x;
  const int lane = tid & 31;
  const int wave = tid >> 5;
  const int half = lane >> 4;
  const int l16  = lane & 15;
  const int h     = blockIdx.y;
  const int qbase = blockIdx.x * 128 + wave * 16;
  const uint32_t ld = 3072;

  const float* qp = qkv + (uint32_t)h * 64u;
  const float* kp = qkv + 1024u + (uint32_t)h * 64u;
  const float* vp = qkv + 2048u + (uint32_t)h * 64u;

  // Preload Q fragments for all 16 K-steps (row-major, K contiguous).
  v2f qf[16];
  {
    const uint32_t qoff = (uint32_t)(qbase + l16) * ld + 2u * half;
#pragma unroll
    for (int kk = 0; kk < 16; ++kk)
      qf[kk] = *(const v2f*)(qp + qoff + 4u * kk);
  }

  v8f o[4];
#pragma unroll
  for (int ct = 0; ct < 4; ++ct) o[ct] = (v8f){};
  float mrow[8], lrow[8];
#pragma unroll
  for (int v = 0; v < 8; ++v) { mrow[v] = -1e30f; lrow[v] = 0.0f; }

  float* myp = Plds + wave * 16 * LDT;
  const int ktmax_w = (qbase + 15) >> 6;              // this wave's causal limit
  const int ktmax_b = (blockIdx.x * 128 + 127) >> 6;  // block-uniform limit

  for (int kt = 0; kt <= ktmax_b; ++kt) {
    const uint32_t tbase = (uint32_t)(kt * 64) * ld;

    __syncthreads();
    // ---- cooperative staging: 64x64 K and V tiles (V transposed) ----
#pragma unroll 4
    for (int t = 0; t < 16; ++t) {
      int idx = t * 256 + tid;
      int key = idx >> 6, c = idx & 63;
      uint32_t goff = tbase + (uint32_t)key * ld + (uint32_t)c;
      float kvv = kp[goff];
      float vvv = vp[goff];
      Klds[key * LDT + c] = kvv;   // contiguous writes: conflict-free
      Vlds[c * LDT + key] = vvv;   // stride-68 writes: banks 4 apart
    }
    // ---- prefetch next tile while this one computes ----
    if (kt < ktmax_b) {
      if (tid < 64) {
        const float* p = kp + tbase + (uint32_t)(64 + tid) * ld;
        __builtin_prefetch(p, 0, 3);
        __builtin_prefetch(p + 32, 0, 3);
      } else if (tid < 128) {
        const float* p = vp + tbase + (uint32_t)tid * ld;  // rows 64..127
        __builtin_prefetch(p, 0, 3);
        __builtin_prefetch(p + 32, 0, 3);
      }
    }
    __syncthreads();

    if (kt <= ktmax_w) {   // wave-uniform: EXEC stays all-ones for WMMA
      // ---- S = Q * K^T over this 64-key tile ----
      v8f s[4];
#pragma unroll
      for (int j = 0; j < 4; ++j) s[j] = (v8f){};
#pragma unroll 4
      for (int kk = 0; kk < 16; ++kk) {
        v2f a = qf[kk];
#pragma unroll
        for (int j = 0; j < 4; ++j) {
          v2f b = *(const v2f*)&Klds[(j * 16 + l16) * LDT + 4 * kk + 2 * half];
          s[j] = wmma_f32(a, b, s[j]);
        }
      }

      // ---- scale, causal mask, online softmax; stage P into padded LDS ----
      float alpha[8];
#pragma unroll
      for (int v = 0; v < 8; ++v) {
        const int row = qbase + half * 8 + v;
        float sv[4];
        float mx = -1e30f;
#pragma unroll
        for (int j = 0; j < 4; ++j) {
          const int col = kt * 64 + j * 16 + l16;
          float x = s[j][v] * 0.125f;          // 1/sqrt(64)
          if (col > row) x = -1e30f;           // causal mask
          sv[j] = x;
          mx = fmaxf(mx, x);
        }
#pragma unroll
        for (int off = 1; off < 16; off <<= 1)  // stays within 16-lane half
          mx = fmaxf(mx, __shfl_xor(mx, off, 32));
        const float mnew = fmaxf(mrow[v], mx);
        const float a = __expf(mrow[v] - mnew);
        float rsum = 0.0f;
#pragma unroll
        for (int j = 0; j < 4; ++j) {
          float p = __expf(sv[j] - mnew);
          myp[(half * 8 + v) * LDT + j * 16 + l16] = p;
          rsum += p;
        }
#pragma unroll
        for (int off = 1; off < 16; off <<= 1)
          rsum += __shfl_xor(rsum, off, 32);
        lrow[v] = lrow[v] * a + rsum;
        mrow[v] = mnew;
        alpha[v] = a;
      }
#pragma unroll
      for (int ct = 0; ct < 4; ++ct)
#pragma unroll
        for (int v = 0; v < 8; ++v) o[ct][v] *= alpha[v];

      // ---- O += P * V  (A from padded P slab, B from transposed V) ----
#pragma unroll 2
      for (int kk = 0; kk < 16; ++kk) {
        v2f a = *(const v2f*)&myp[l16 * LDT + 4 * kk + 2 * half];
#pragma unroll
        for (int ct = 0; ct < 4; ++ct) {
          v2f b = *(const v2f*)&Vlds[(ct * 16 + l16) * LDT + 4 * kk + 2 * half];
          o[ct] = wmma_f32(a, b, o[ct]);
        }
      }
    }
  }

  // ---- epilogue: O / l, write [T x 1024] row-major ----
#pragma unroll
  for (int ct = 0; ct < 4; ++ct)
#pragma unroll
    for (int v = 0; v < 8; ++v) {
      const uint32_t row = (uint32_t)(qbase + half * 8 + v);
      out[row * 1024u + (uint32_t)(h * 64 + ct * 16 + l16)] =
          o[ct][v] / lrow[v];
    }
}

// ---------------------------------------------------------------------------
// Launch: qkv GEMM -> RoPE -> flash attention -> output GEMM
// Workspace: qkv (4096*3072 f32 = 48 MB) + attnout (4096*1024 f32 = 16 MB)
// ---------------------------------------------------------------------------
extern "C" void kernel_launch(void* const* d_in, const int* in_sizes, int n_in,
                              void* d_out, int out_size, void* d_ws, size_t ws_size,
                              hipStream_t stream) {
  const float* x     = (const float*)d_in[0];   // [4096, 1024]
  const float* Wqkv  = (const float*)d_in[1];   // [3072, 1024]
  const float* Wproj = (const float*)d_in[2];   // [1024, 1024]
  float* out = (float*)d_out;                   // [4096, 1024]

  float* qkv     = (float*)d_ws;                 // [4096, 3072]
  float* attnout = qkv + (size_t)4096 * 3072;    // [4096, 1024]

  // qkv = x @ Wqkv^T : M=4096, N=3072, K=1024
  gemm_nt_wmma<<<dim3(3072 / 64, 4096 / 32), 32, 0, stream>>>(
      x, Wqkv, qkv, 1024, 1024, 1024, 3072);

  // RoPE on q,k halves in place
  rope_kernel<<<(4096 * 1024) / 256, 256, 0, stream>>>(qkv, 4096);

  // Flash attention: grid (32 q-tiles of 128 rows, 16 heads)
  attn_kernel<<<dim3(32, 16), 256, 0, stream>>>(qkv, attnout);

  // out = attnout @ Wproj^T : M=4096, N=1024, K=1024
  gemm_nt_wmma<<<dim3(1024 / 64, 4096 / 32), 32, 0, stream>>>(
      attnout, Wproj, out, 1024, 1024, 1024, 1024);
}